// DGMC_80788334838138
// MI455X (gfx1250) — compile-verified
//
#include <hip/hip_runtime.h>
#include <hip/hip_bf16.h>
#include <math.h>

// Problem constants (match reference)
#define BB    4
#define NS    512
#define NT    512
#define D_IN  128
#define C1    64
#define RR    32
#define C2    32
#define EE    32768
#define STEPS 10

typedef float v2f __attribute__((ext_vector_type(2)));
typedef float v8f __attribute__((ext_vector_type(8)));

// ---------------------------------------------------------------------------
// Zero-fill
// ---------------------------------------------------------------------------
__global__ void zero_f32(float* __restrict__ p, long n) {
    long i = (long)blockIdx.x * blockDim.x + threadIdx.x;
    if (i < n) p[i] = 0.0f;
}

// ---------------------------------------------------------------------------
// Edge scatter: m[dst*F + f] += x[src*F + f] * ea[e]
// ei layout: ei[0..E) = src, ei[E..2E) = dst
// ---------------------------------------------------------------------------
__global__ void edge_scatter(const float* __restrict__ x, const int* __restrict__ ei,
                             const float* __restrict__ ea, float* __restrict__ m,
                             int E, int F) {
    long idx = (long)blockIdx.x * blockDim.x + threadIdx.x;
    long total = (long)E * F;
    if (idx >= total) return;
    int e = (int)(idx / F);
    int f = (int)(idx % F);
    int src = ei[e];
    int dst = ei[E + e];
    atomicAdd(&m[(long)dst * F + f], x[(long)src * F + f] * ea[e]);
}

// ---------------------------------------------------------------------------
// WMMA f32 GEMM, compile-time specialized:
//   C = maybe_relu( op(A)[+Aadd] * op(B) + bias )
//   - one wave per 16x16 output tile, K in steps of 4 via v_wmma_f32_16x16x4_f32
//   - TA: A is K x M (A[k*lda + m]); else A is M x K (A[m*lda + k])
//   - TB: B is N x K (B[n*ldb + k]); else B is K x N (B[k*ldb + n])
//   - HASADD: elementwise addend for A (non-trans layout; only with TA=false)
//   - HASBIAS: per-output-column bias
//   - blockIdx.z batches with strideA/B/C
// A-fragment (16x4 f32): lane<16 -> M=lane holds K={0,1}; lane>=16 -> K={2,3}
// B-fragment (4x16 f32): mirrored on N. C/D: 8 VGPRs, row = v + 8*(lane>>4),
// col = lane&15.
// ---------------------------------------------------------------------------
template <bool TA, bool TB, bool HASADD, bool HASBIAS, bool RELU>
__global__ void wmma_gemm_f32(const float* __restrict__ A,
                              const float* __restrict__ Aadd,
                              const float* __restrict__ Bm,
                              float* __restrict__ C,
                              const float* __restrict__ bias,
                              int M, int N, int K,
                              int lda, int ldb, int ldc,
                              long strideA, long strideB, long strideC) {
    const int lane = threadIdx.x & 31;
    const int wave = threadIdx.x >> 5;
    const int tilesN = N >> 4;
    const int tilesM = M >> 4;
    const int tile = blockIdx.x * (blockDim.x >> 5) + wave;
    if (tile >= tilesM * tilesN) return;   // uniform per wave -> EXEC stays all-ones

    const int tm = (tile / tilesN) << 4;
    const int tn = (tile % tilesN) << 4;

    const float* Ab = A  + (long)blockIdx.z * strideA;
    const float* Bb = Bm + (long)blockIdx.z * strideB;
    float*       Cb = C  + (long)blockIdx.z * strideC;

    const int r16 = lane & 15;          // M row of A-frag == N col of B-frag
    const int kb  = (lane >> 4) << 1;   // 0 or 2 : K pair owned by this half-wave

    // Incremental per-lane pointers (no per-iteration index math)
    const float* pA = TA ? (Ab + (long)kb * lda + (tm + r16))
                         : (Ab + (long)(tm + r16) * lda + kb);
    const float* pAd = HASADD ? (Aadd + (long)(tm + r16) * lda + kb) : nullptr;
    const float* pB = TB ? (Bb + (long)(tn + r16) * ldb + kb)
                         : (Bb + (long)kb * ldb + (tn + r16));
    const long stepA = TA ? (long)4 * lda : 4;
    const long stepB = TB ? 4 : (long)4 * ldb;

    v8f acc = {0.f, 0.f, 0.f, 0.f, 0.f, 0.f, 0.f, 0.f};

#pragma unroll 4
    for (int k = 0; k < K; k += 4) {
        v2f a, b;
        if (TA) {                       // column access, stride lda between ks
            a.x = pA[0];
            a.y = pA[lda];
        } else {                        // contiguous pair -> global_load_b64
            a = *(const v2f*)pA;
            if (HASADD) {
                v2f q = *(const v2f*)pAd;
                a.x += q.x; a.y += q.y;
                pAd += 4;
            }
        }
        if (TB) {                       // contiguous pair -> global_load_b64
            b = *(const v2f*)pB;
        } else {                        // row access, stride ldb between ks
            b.x = pB[0];
            b.y = pB[ldb];
        }
        acc = __builtin_amdgcn_wmma_f32_16x16x4_f32(
            /*neg_a=*/false, a, /*neg_b=*/false, b,
            /*c_mod=*/(short)0, acc, /*reuse_a=*/false, /*reuse_b=*/false);
        pA += stepA;
        pB += stepB;
    }

    const int col = tn + r16;
    float bv = 0.0f;
    if (HASBIAS) bv = bias[col];
    const int rbase = tm + ((lane >> 4) << 3);
#pragma unroll
    for (int v = 0; v < 8; ++v) {
        float val = acc[v] + bv;
        if (RELU) val = fmaxf(val, 0.0f);
        Cb[(long)(rbase + v) * ldc + col] = val;
    }
}

// ---------------------------------------------------------------------------
// Row softmax over `cols` elements; one block per row. Writes S, and optionally
// duplicates the result into `extra` (used for S_0 / S_L outputs).
// ---------------------------------------------------------------------------
__global__ void softmax_rows(const float* __restrict__ Shat, float* __restrict__ S,
                             float* __restrict__ extra, int cols) {
    __shared__ float red[256];
    const int row = blockIdx.x;
    const float* in = Shat + (long)row * cols;

    float lmax = -INFINITY;
    for (int c = threadIdx.x; c < cols; c += blockDim.x)
        lmax = fmaxf(lmax, in[c]);
    red[threadIdx.x] = lmax;
    __syncthreads();
    for (int s = 128; s > 0; s >>= 1) {
        if (threadIdx.x < s) red[threadIdx.x] = fmaxf(red[threadIdx.x], red[threadIdx.x + s]);
        __syncthreads();
    }
    const float rmax = red[0];
    __syncthreads();

    float lsum = 0.0f;
    for (int c = threadIdx.x; c < cols; c += blockDim.x)
        lsum += __expf(in[c] - rmax);
    red[threadIdx.x] = lsum;
    __syncthreads();
    for (int s = 128; s > 0; s >>= 1) {
        if (threadIdx.x < s) red[threadIdx.x] += red[threadIdx.x + s];
        __syncthreads();
    }
    const float inv = 1.0f / red[0];

    for (int c = threadIdx.x; c < cols; c += blockDim.x) {
        float v = __expf(in[c] - rmax) * inv;
        S[(long)row * cols + c] = v;
        if (extra) extra[(long)row * cols + c] = v;
    }
}

// ---------------------------------------------------------------------------
// Consensus update:
//   S_hat[b,s,t] += mb2 + sum_k mw2[k] * relu(a_s[b,s,k] - a_t[b,t,k])
// (mb1 already folded into a_s). 16x16 (s,t) tile per block, a_s/a_t in LDS.
// ---------------------------------------------------------------------------
__global__ void consensus_update(float* __restrict__ Shat,
                                 const float* __restrict__ a_s,
                                 const float* __restrict__ a_t,
                                 const float* __restrict__ mw2,
                                 const float* __restrict__ mb2) {
    __shared__ float As[16][C2];
    __shared__ float At[16][C2];
    __shared__ float w[C2];

    const int b  = blockIdx.z;
    const int s0 = blockIdx.y << 4;
    const int t0 = blockIdx.x << 4;
    const int tid = threadIdx.x;

    {   // 256 threads load 16x32 of a_s and a_t (2 floats each)
        const int r  = tid >> 4;
        const int c2 = (tid & 15) << 1;
        const float* ps = a_s + ((long)(b * NS + s0 + r) * C2 + c2);
        As[r][c2] = ps[0]; As[r][c2 + 1] = ps[1];
        const float* pt = a_t + ((long)(b * NT + t0 + r) * C2 + c2);
        At[r][c2] = pt[0]; At[r][c2 + 1] = pt[1];
    }
    if (tid < C2) w[tid] = mw2[tid];
    __syncthreads();

    const int ts = tid >> 4;
    const int tt = tid & 15;
    float acc = mb2[0];
#pragma unroll
    for (int k = 0; k < C2; ++k)
        acc += w[k] * fmaxf(As[ts][k] - At[tt][k], 0.0f);

    Shat[((long)(b * NS + s0 + ts)) * NT + (t0 + tt)] += acc;
}

// ---------------------------------------------------------------------------
// Host-side helpers
// ---------------------------------------------------------------------------
static inline void run_zero(float* p, long n, hipStream_t st) {
    zero_f32<<<(unsigned)((n + 255) / 256), 256, 0, st>>>(p, n);
}

template <bool TA, bool TB, bool HASADD, bool HASBIAS, bool RELU>
static inline void run_gemm(const float* A, const float* Aadd, const float* Bm, float* C,
                            const float* bias, int M, int N, int K,
                            int lda, int ldb, int ldc,
                            long sA, long sB, long sC, int nbatch, hipStream_t st) {
    const int tiles = (M >> 4) * (N >> 4);
    dim3 grid((tiles + 7) / 8, 1, nbatch);   // 8 waves (16x16 tiles) per 256-thread block
    wmma_gemm_f32<TA, TB, HASADD, HASBIAS, RELU><<<grid, 256, 0, st>>>(
        A, Aadd, Bm, C, bias, M, N, K, lda, ldb, ldc, sA, sB, sC);
}

extern "C" void kernel_launch(void* const* d_in, const int* in_sizes, int n_in,
                              void* d_out, int out_size, void* d_ws, size_t ws_size,
                              hipStream_t stream) {
    (void)in_sizes; (void)n_in; (void)out_size; (void)ws_size;

    const float* x_s  = (const float*)d_in[0];
    const int*   ei_s = (const int*)  d_in[1];
    const float* ea_s = (const float*)d_in[2];
    const float* x_t  = (const float*)d_in[4];
    const int*   ei_t = (const int*)  d_in[5];
    const float* ea_t = (const float*)d_in[6];
    const float* W1   = (const float*)d_in[8];    // (128,64)
    const float* W2   = (const float*)d_in[9];    // (32,32)
    const float* mw1  = (const float*)d_in[10];   // (32,32)
    const float* mb1  = (const float*)d_in[11];   // (32,)
    const float* mw2  = (const float*)d_in[12];   // (32,1)
    const float* mb2  = (const float*)d_in[13];   // (1,)
    const float* r_all = (const float*)d_in[14];  // (10,4,512,32)

    float* out = (float*)d_out;
    float* S0_out = out;                              // [4,512,512]
    float* SL_out = out + (long)BB * NS * NT;         // [4,512,512]

    // Workspace carve-up (floats)
    float* W = (float*)d_ws;
    float* m_s  = W;                      // 2048*128 (reused as 2048*32 in the loop)
    float* m_t  = m_s  + (long)BB * NS * D_IN;
    float* h_s  = m_t  + (long)BB * NT * D_IN;   // 2048*64
    float* h_t  = h_s  + (long)BB * NS * C1;
    float* Shat = h_t  + (long)BB * NT * C1;     // 4*512*512
    float* Smat = Shat + (long)BB * NS * NT;     // 4*512*512
    float* r_t  = Smat + (long)BB * NS * NT;     // 4*512*32
    float* o_s  = r_t  + (long)BB * NT * RR;     // 2048*32
    float* o_t  = o_s  + (long)BB * NS * C2;
    float* a_s  = o_t  + (long)BB * NT * C2;
    float* a_t  = a_s  + (long)BB * NS * C2;

    const long nNodes = (long)BB * NS;           // 2048 (== B*NT too)

    // ---- psi with W1: h = relu((x + segsum) @ W1) ----
    run_zero(m_s, nNodes * D_IN, stream);
    run_zero(m_t, nNodes * D_IN, stream);
    {
        long tot = (long)EE * D_IN;
        edge_scatter<<<(unsigned)((tot + 255) / 256), 256, 0, stream>>>(x_s, ei_s, ea_s, m_s, EE, D_IN);
        edge_scatter<<<(unsigned)((tot + 255) / 256), 256, 0, stream>>>(x_t, ei_t, ea_t, m_t, EE, D_IN);
    }
    // <TA,TB,HASADD,HASBIAS,RELU>
    run_gemm<false, false, true, false, true>(x_s, m_s, W1, h_s, nullptr,
        (int)nNodes, C1, D_IN, D_IN, C1, C1, 0, 0, 0, 1, stream);
    run_gemm<false, false, true, false, true>(x_t, m_t, W1, h_t, nullptr,
        (int)nNodes, C1, D_IN, D_IN, C1, C1, 0, 0, 0, 1, stream);

    // ---- S_hat0[b] = h_s[b] @ h_t[b]^T  (batched, transB) ----
    run_gemm<false, true, false, false, false>(h_s, nullptr, h_t, Shat, nullptr,
        NS, NT, C1, C1, C1, NT,
        (long)NS * C1, (long)NT * C1, (long)NS * NT, BB, stream);

    // ---- S_0 = softmax(S_hat0) ; also seeds the loop's S ----
    softmax_rows<<<BB * NS, 256, 0, stream>>>(Shat, Smat, S0_out, NT);

    // ---- 10 consensus steps ----
    for (int i = 0; i < STEPS; ++i) {
        const float* ri = r_all + (long)i * BB * NS * RR;

        // r_t[b] = S[b]^T @ r_i[b]   (batched, transA)
        run_gemm<true, false, false, false, false>(Smat, nullptr, ri, r_t, nullptr,
            NT, RR, NS, NT, RR, RR,
            (long)NS * NT, (long)NS * RR, (long)NT * RR, BB, stream);

        // o_s = relu((r_i + segsum_s) @ W2)
        run_zero(m_s, nNodes * RR, stream);
        {
            long tot = (long)EE * RR;
            edge_scatter<<<(unsigned)((tot + 255) / 256), 256, 0, stream>>>(ri, ei_s, ea_s, m_s, EE, RR);
        }
        run_gemm<false, false, true, false, true>(ri, m_s, W2, o_s, nullptr,
            (int)nNodes, C2, RR, RR, C2, C2, 0, 0, 0, 1, stream);

        // o_t = relu((r_t + segsum_t) @ W2)
        run_zero(m_t, nNodes * RR, stream);
        {
            long tot = (long)EE * RR;
            edge_scatter<<<(unsigned)((tot + 255) / 256), 256, 0, stream>>>(r_t, ei_t, ea_t, m_t, EE, RR);
        }
        run_gemm<false, false, true, false, true>(r_t, m_t, W2, o_t, nullptr,
            (int)nNodes, C2, RR, RR, C2, C2, 0, 0, 0, 1, stream);

        // a_s = o_s @ mw1 + mb1 ; a_t = o_t @ mw1   (mb1 folded into s side)
        run_gemm<false, false, false, true, false>(o_s, nullptr, mw1, a_s, mb1,
            (int)nNodes, C2, C2, C2, C2, C2, 0, 0, 0, 1, stream);
        run_gemm<false, false, false, false, false>(o_t, nullptr, mw1, a_t, nullptr,
            (int)nNodes, C2, C2, C2, C2, C2, 0, 0, 0, 1, stream);

        // S_hat += mb2 + sum_k mw2[k]*relu(a_s - a_t)
        {
            dim3 grid(NT / 16, NS / 16, BB);
            consensus_update<<<grid, 256, 0, stream>>>(Shat, a_s, a_t, mw2, mb2);
        }

        // refresh S for next step; last iteration also emits S_L
        softmax_rows<<<BB * NS, 256, 0, stream>>>(Shat, Smat,
                                                  (i == STEPS - 1) ? SL_out : nullptr, NT);
    }
}